// GraphFusionNetwork_36945308680269
// MI455X (gfx1250) — compile-verified
//
#include <hip/hip_runtime.h>
#include <hip/hip_bf16.h>

typedef __attribute__((ext_vector_type(2))) float v2f;
typedef __attribute__((ext_vector_type(8))) float v8f;

constexpr int G  = 4;
constexpr int N  = 100000;
constexpr int E  = 500000;
constexpr int B  = 512;
constexpr int D  = 300;
constexpr int H  = 128;
constexpr int C  = 20;
constexpr int NH = 3;

constexpr size_t GNH = (size_t)G * N * H;   // 51.2M floats
constexpr size_t GE  = (size_t)G * E;       // 2M floats
constexpr size_t GN  = (size_t)G * N;       // 400k floats
constexpr size_t GBH = (size_t)G * B * H;   // 262k floats
constexpr size_t GB  = (size_t)G * B;       // 2048 floats

// ---------------------------------------------------------------------------
// 1) edge weights w = relu(ep*ew) and in-degree counts (atomic)
// ---------------------------------------------------------------------------
__global__ void k_edge_w_deg(const float* __restrict__ ew, const float* __restrict__ ep,
                             const int* __restrict__ dst, float* __restrict__ wbuf,
                             float* __restrict__ deg) {
    size_t i = (size_t)blockIdx.x * blockDim.x + threadIdx.x;
    if (i >= GE) return;
    int g = (int)(i / E);
    float w = ep[g] * ew[i];
    wbuf[i] = w > 0.f ? w : 0.f;
    atomicAdd(deg + (size_t)g * N + dst[i], 1.0f);
}

// ---------------------------------------------------------------------------
// 2) inverse degree
// ---------------------------------------------------------------------------
__global__ void k_invdeg(const float* __restrict__ deg, float* __restrict__ inv) {
    size_t i = (size_t)blockIdx.x * blockDim.x + threadIdx.x;
    if (i >= GN) return;
    float d = deg[i];
    inv[i] = d > 0.f ? 1.0f / d : 0.0f;
}

// ---------------------------------------------------------------------------
// 3) projection: h0[g,n,:] = emb_table[idx[g,n],:] @ W_proj[g] + b_proj[g]
//    fp32 WMMA 16x16x4, 8 column-blocks (128 cols), K=300 via 3 LDS panels of 100.
//    Panels staged with GLOBAL_LOAD_ASYNC_TO_LDS_B128 (ASYNCcnt-tracked DMA,
//    no VGPR round-trip). Block = 256 threads = 8 waves; one 16-row tile/wave.
// ---------------------------------------------------------------------------
__global__ void __launch_bounds__(256)
k_proj(const int* __restrict__ node_indices, const float* __restrict__ emb,
       const float* __restrict__ Wp, const float* __restrict__ bp,
       float* __restrict__ hOut) {
    __shared__ float ldsW[100 * H];             // 51.2 KB K-panel of W_proj[g]

    const int g    = blockIdx.y;
    const int tid  = threadIdx.x;
    const int wave = tid >> 5;
    const int lane = tid & 31;
    const int l16  = lane & 15;
    const bool hi  = lane >= 16;                // lanes 16..31 carry K+2,K+3
    const int tileIdx = blockIdx.x * 8 + wave;
    const bool active = tileIdx < (N / 16);     // 6250 tiles exactly cover N
    const int rowBase = tileIdx * 16;

    v8f acc[8];
    #pragma unroll
    for (int nb = 0; nb < 8; ++nb)
        #pragma unroll
        for (int j = 0; j < 8; ++j) acc[nb][j] = 0.0f;

    const float* WpG = Wp + (size_t)g * D * H;

    // LDS byte offset of ldsW (low 32 bits of an LDS generic address = offset)
    const unsigned ldsBase = (unsigned)(uintptr_t)(void*)ldsW;

    // gather base for this lane's A row (A 16x4 layout: lane holds M=l16)
    int vidx = 0;
    if (active) vidx = node_indices[(size_t)g * N + rowBase + l16];
    const float* aBase = emb + (size_t)vidx * D + (hi ? 2 : 0);

    for (int kp = 0; kp < 3; ++kp) {            // 3 K-panels of 100
        __syncthreads();                        // prior panel reads complete
        {   // async DMA panel copy: 100*128 floats = 3200 x b128
            const char* gsrc = (const char*)(WpG + kp * 100 * H);
            for (int i = tid; i < 100 * H / 4; i += 256) {
                unsigned lds_addr = ldsBase + (unsigned)(i * 16);
                unsigned long long gaddr =
                    (unsigned long long)(uintptr_t)(gsrc + (size_t)i * 16);
                asm volatile("global_load_async_to_lds_b128 %0, %1, off"
                             :: "v"(lds_addr), "v"(gaddr) : "memory");
            }
            asm volatile("s_wait_asynccnt 0x0" ::: "memory");
        }
        __syncthreads();                        // panel visible to all waves
        if (active) {
            #pragma unroll 1
            for (int kk = 0; kk < 100; kk += 4) {
                // A tile: lane<16 -> (K+0,K+1), lane>=16 -> (K+2,K+3), row M=l16
                v2f a = *reinterpret_cast<const v2f*>(aBase + kp * 100 + kk);
                const float* bptr = ldsW + (kk + (hi ? 2 : 0)) * H + l16;
                #pragma unroll
                for (int nb = 0; nb < 8; ++nb) {
                    // B tile: VGPR0 = row K(+0/+2), VGPR1 = row K(+1/+3), col N=l16
                    v2f b;
                    b.x = bptr[nb * 16];
                    b.y = bptr[nb * 16 + H];
                    acc[nb] = __builtin_amdgcn_wmma_f32_16x16x4_f32(
                        false, a, false, b, (short)0, acc[nb], false, false);
                }
            }
        }
    }

    if (active) {
        #pragma unroll
        for (int nb = 0; nb < 8; ++nb) {
            int col = nb * 16 + l16;
            float bias = bp[(size_t)g * H + col];
            int rBase = rowBase + (hi ? 8 : 0);   // D layout: lane>=16 -> M+8
            #pragma unroll
            for (int v = 0; v < 8; ++v)
                hOut[((size_t)g * N + rBase + v) * H + col] = acc[nb][v] + bias;
        }
    }
}

// ---------------------------------------------------------------------------
// 4) message scatter: hNext[dst] += hCur[src] * w   (one wave per edge, f32 atomics)
// ---------------------------------------------------------------------------
__global__ void __launch_bounds__(256)
k_scatter(const int* __restrict__ src, const int* __restrict__ dst,
          const float* __restrict__ wbuf, const float* __restrict__ hCur,
          float* __restrict__ hNext) {
    const int lane = threadIdx.x & 31;
    size_t eIdx = (size_t)blockIdx.x * 8 + (threadIdx.x >> 5);
    if (eIdx >= GE) return;
    int g = (int)(eIdx / E);
    int s = src[eIdx];
    int d = dst[eIdx];
    float w = wbuf[eIdx];
    float4 hv = *reinterpret_cast<const float4*>(hCur + ((size_t)g * N + s) * H + lane * 4);
    float* out = hNext + ((size_t)g * N + d) * H + lane * 4;
    atomicAdd(out + 0, hv.x * w);
    atomicAdd(out + 1, hv.y * w);
    atomicAdd(out + 2, hv.z * w);
    atomicAdd(out + 3, hv.w * w);
}

// ---------------------------------------------------------------------------
// 5) combine: hNext = hCur + hNext * inv_deg   (vectorized)
// ---------------------------------------------------------------------------
__global__ void k_combine(const float* __restrict__ hCur, float* __restrict__ hNext,
                          const float* __restrict__ inv) {
    size_t i4 = (size_t)blockIdx.x * blockDim.x + threadIdx.x;
    if (i4 >= GNH / 4) return;
    size_t node = i4 / (H / 4);                 // flat g*N + n
    float s = inv[node];
    float4 c = reinterpret_cast<const float4*>(hCur)[i4];
    float4 a = reinterpret_cast<float4*>(hNext)[i4];
    a.x = c.x + a.x * s;
    a.y = c.y + a.y * s;
    a.z = c.z + a.z * s;
    a.w = c.w + a.w * s;
    reinterpret_cast<float4*>(hNext)[i4] = a;
}

// ---------------------------------------------------------------------------
// 6) per-document sums (atomic scatter): one block (128 thr) per node
// ---------------------------------------------------------------------------
__global__ void k_doc(const int* __restrict__ gid, const float* __restrict__ h,
                      float* __restrict__ docSum, float* __restrict__ docCnt) {
    int n = blockIdx.x % N;
    int g = blockIdx.x / N;
    int t = threadIdx.x;                        // 0..127
    int b = gid[(size_t)g * N + n];
    atomicAdd(docSum + ((size_t)g * B + b) * H + t, h[((size_t)g * N + n) * H + t]);
    if (t == 0) atomicAdd(docCnt + (size_t)g * B + b, 1.0f);
}

// ---------------------------------------------------------------------------
// 7) classify: logits[g,b,:] = relu(doc) @ W_cls[g] + b_cls[g]
// ---------------------------------------------------------------------------
__global__ void k_classify(const float* __restrict__ docSum, const float* __restrict__ docCnt,
                           const float* __restrict__ Wc, const float* __restrict__ bc,
                           float* __restrict__ logits) {
    __shared__ float sdoc[H];
    int b = blockIdx.x % B;
    int g = blockIdx.x / B;
    int t = threadIdx.x;                        // 0..127
    float cnt = docCnt[(size_t)g * B + b];
    cnt = cnt > 1.0f ? cnt : 1.0f;
    float v = docSum[((size_t)g * B + b) * H + t] / cnt;
    sdoc[t] = v > 0.0f ? v : 0.0f;
    __syncthreads();
    if (t < C) {
        float acc = bc[(size_t)g * C + t];
        for (int hh = 0; hh < H; ++hh)
            acc += sdoc[hh] * Wc[((size_t)g * H + hh) * C + t];
        logits[((size_t)g * B + b) * C + t] = acc;
    }
}

// ---------------------------------------------------------------------------
// 8) fuse heads (Conv1d k=G per head, mean over heads) + softmax
// ---------------------------------------------------------------------------
__global__ void k_fuse(const float* __restrict__ logits, const float* __restrict__ Wf,
                       const float* __restrict__ bf, float* __restrict__ outPred,
                       float* __restrict__ outLogit) {
    __shared__ float sl[G * C];
    __shared__ float sf[C];
    int b = blockIdx.x;
    int t = threadIdx.x;                        // 0..31 (one wave)
    for (int i = t; i < G * C; i += 32) {
        int g = i / C, ii = i % C;
        sl[i] = logits[((size_t)g * B + b) * C + ii];   // stacked[b][ii][g]
    }
    __syncthreads();
    if (t < C) {
        float acc = 0.0f;
        for (int h2 = 0; h2 < NH; ++h2) {
            float ha = bf[h2 * C + t];
            for (int i = 0; i < C; ++i) {
                const float* wrow = Wf + (((size_t)h2 * C + t) * C + i) * G;
                for (int g = 0; g < G; ++g) ha += sl[g * C + i] * wrow[g];
            }
            acc += ha;
        }
        sf[t] = acc * (1.0f / NH);
    }
    __syncthreads();
    if (t < C) {
        float mx = -3.402823e38f;
        for (int i = 0; i < C; ++i) mx = fmaxf(mx, sf[i]);
        float sum = 0.0f;
        for (int i = 0; i < C; ++i) sum += __expf(sf[i] - mx);
        outPred[(size_t)b * C + t]  = __expf(sf[t] - mx) / sum;
        outLogit[(size_t)b * C + t] = sf[t];
    }
}

// ---------------------------------------------------------------------------
extern "C" void kernel_launch(void* const* d_in, const int* in_sizes, int n_in,
                              void* d_out, int out_size, void* d_ws, size_t ws_size,
                              hipStream_t stream) {
    const int*   node_indices = (const int*)d_in[0];
    const int*   src          = (const int*)d_in[1];
    const int*   dst          = (const int*)d_in[2];
    const int*   gid          = (const int*)d_in[3];
    const float* ew           = (const float*)d_in[4];
    const float* emb          = (const float*)d_in[5];
    const float* ep           = (const float*)d_in[6];
    const float* Wp           = (const float*)d_in[7];
    const float* bp           = (const float*)d_in[8];
    const float* Wc           = (const float*)d_in[9];
    const float* bc           = (const float*)d_in[10];
    const float* Wf           = (const float*)d_in[11];
    const float* bf           = (const float*)d_in[12];
    float* out = (float*)d_out;

    float* ws      = (float*)d_ws;
    float* hA      = ws;                  // [G,N,H]
    float* hB      = hA + GNH;            // [G,N,H]
    float* wbuf    = hB + GNH;            // [G,E]
    float* invdeg  = wbuf + GE;           // [G,N]
    float* deg     = invdeg + GN;         // [G,N]
    float* docSum  = deg + GN;            // [G,B,H]
    float* docCnt  = docSum + GBH;        // [G,B]
    float* logits  = docCnt + GB;         // [G,B,C]

    // zero deg + docSum + docCnt (contiguous)
    hipMemsetAsync(deg, 0, (GN + GBH + GB) * sizeof(float), stream);

    k_edge_w_deg<<<(int)((GE + 255) / 256), 256, 0, stream>>>(ew, ep, dst, wbuf, deg);
    k_invdeg<<<(int)((GN + 255) / 256), 256, 0, stream>>>(deg, invdeg);

    dim3 pgrid((N / 16 + 7) / 8, G);      // 782 x 4 blocks, 8 waves each
    k_proj<<<pgrid, 256, 0, stream>>>(node_indices, emb, Wp, bp, hA);

    // step 1: hB = hA + scatter(hA)*inv
    hipMemsetAsync(hB, 0, GNH * sizeof(float), stream);
    k_scatter<<<(int)((GE + 7) / 8), 256, 0, stream>>>(src, dst, wbuf, hA, hB);
    k_combine<<<(int)((GNH / 4 + 255) / 256), 256, 0, stream>>>(hA, hB, invdeg);

    // step 2: hA = hB + scatter(hB)*inv
    hipMemsetAsync(hA, 0, GNH * sizeof(float), stream);
    k_scatter<<<(int)((GE + 7) / 8), 256, 0, stream>>>(src, dst, wbuf, hB, hA);
    k_combine<<<(int)((GNH / 4 + 255) / 256), 256, 0, stream>>>(hB, hA, invdeg);

    k_doc<<<G * N, H, 0, stream>>>(gid, hA, docSum, docCnt);
    k_classify<<<G * B, H, 0, stream>>>(docSum, docCnt, Wc, bc, logits);
    k_fuse<<<B, 32, 0, stream>>>(logits, Wf, bf, out, out + (size_t)B * C);
}